// Solver_61632780698038
// MI455X (gfx1250) — compile-verified
//
#include <hip/hip_runtime.h>
#include <math.h>

// Problem constants (match reference).
#define Bn    4
#define Hn    480
#define Wn    640
#define NOFFn 8
#define HWn   (Hn * Wn)
#define Npx   (Bn * HWn)

// 3D launch: W = 10*64, H = 120*4 -> no integer divisions anywhere,
// b = blockIdx.z is wave-uniform so K loads scalarize.
#define BX 64
#define BY 4

__device__ __forceinline__ void kinv3(const float* __restrict__ Kp, float inv[9]) {
    const float m00 = Kp[0], m01 = Kp[1], m02 = Kp[2];
    const float m10 = Kp[3], m11 = Kp[4], m12 = Kp[5];
    const float m20 = Kp[6], m21 = Kp[7], m22 = Kp[8];
    const float c00 = m11 * m22 - m12 * m21;
    const float c10 = m12 * m20 - m10 * m22;
    const float c20 = m10 * m21 - m11 * m20;
    const float det = m00 * c00 + m01 * c10 + m02 * c20;
    const float id  = 1.0f / det;
    inv[0] = c00 * id;
    inv[1] = (m02 * m21 - m01 * m22) * id;
    inv[2] = (m01 * m12 - m02 * m11) * id;
    inv[3] = c10 * id;
    inv[4] = (m00 * m22 - m02 * m20) * id;
    inv[5] = (m02 * m10 - m00 * m12) * id;
    inv[6] = c20 * id;
    inv[7] = (m01 * m20 - m00 * m21) * id;
    inv[8] = (m00 * m11 - m01 * m10) * id;
}

// ---------------------------------------------------------------------------
// Pass 0: pack per-pixel gather payloads.
//   packA = { nom = depth*g,  g = n.Kinv[x,y,1],  a = n.Kinv_col0,  b = n.Kinv_col1 }
//   packB = { conf*confN, R, G, B }
// Optionally (OFFP) also interleave the 8 offset pairs into [k][p] float2 planes
// so every gather corner needs one b64 instead of two b32 loads 1.2MB apart.
// ---------------------------------------------------------------------------
template <bool OFFP>
__global__ void __launch_bounds__(BX * BY)
pack_kernel(const float* __restrict__ depth, const float* __restrict__ normal,
            const float* __restrict__ image, const float* __restrict__ conf,
            const float* __restrict__ confN, const float* __restrict__ K,
            const float* __restrict__ offs,
            float4* __restrict__ packA, float4* __restrict__ packB,
            float2* __restrict__ offp) {
    const int x   = blockIdx.x * BX + threadIdx.x;
    const int y   = blockIdx.y * BY + threadIdx.y;
    const int b   = blockIdx.z;                 // wave-uniform
    const int rem = y * Wn + x;
    const int p   = b * HWn + rem;

    float inv[9];
    kinv3(K + b * 9, inv);                       // scalar loads (uniform b)
    const float fx = (float)x, fy = (float)y;
    const float v0 = fmaf(inv[0], fx, fmaf(inv[1], fy, inv[2]));
    const float v1 = fmaf(inv[3], fx, fmaf(inv[4], fy, inv[5]));
    const float v2 = fmaf(inv[6], fx, fmaf(inv[7], fy, inv[8]));

    const int nb = b * 3 * HWn + rem;
    const float nx0 = normal[nb], ny0 = normal[nb + HWn], nz0 = normal[nb + 2 * HWn];

    const float g  = fmaf(nx0, v0, fmaf(ny0, v1, nz0 * v2));
    const float a  = fmaf(nx0, inv[0], fmaf(ny0, inv[3], nz0 * inv[6]));
    const float bb = fmaf(nx0, inv[1], fmaf(ny0, inv[4], nz0 * inv[7]));

    packA[p] = make_float4(depth[p] * g, g, a, bb);
    packB[p] = make_float4(conf[p] * confN[p],
                           image[nb], image[nb + HWn], image[nb + 2 * HWn]);

    if (OFFP) {
        const float* __restrict__ ob = offs + (size_t)b * 2 * NOFFn * HWn + rem;
#pragma unroll
        for (int k = 0; k < NOFFn; ++k)
            offp[(size_t)k * Npx + p] =
                make_float2(ob[(2 * k) * HWn], ob[(2 * k + 1) * HWn]);
    }
}

__device__ __forceinline__ float2 load_off(const float* __restrict__ offbase,
                                           const float2* __restrict__ offp,
                                           bool useOffp, int k, int pGlobal, int rel) {
    if (useOffp) return offp[(size_t)k * Npx + pGlobal];
    return make_float2(offbase[(2 * k) * HWn + rel],
                       offbase[(2 * k + 1) * HWn + rel]);
}

// ---------------------------------------------------------------------------
// Pass 1: bilinear gather of {d1_p, conf*confN, rgb, conf}; cross-bilateral
// depth fusion -> updated_depth (into d_out), cdconf_k -> cw ([k][p] layout).
// ---------------------------------------------------------------------------
template <bool OFFP>
__global__ void __launch_bounds__(BX * BY)
pass1_kernel(const float* __restrict__ depth, const float* __restrict__ image,
             const float* __restrict__ conf, const float* __restrict__ offs,
             const float2* __restrict__ offp,
             const float4* __restrict__ packA, const float4* __restrict__ packB,
             float* __restrict__ cw, float* __restrict__ ud_out) {
    const int x    = blockIdx.x * BX + threadIdx.x;
    const int y    = blockIdx.y * BY + threadIdx.y;
    const int b    = blockIdx.z;
    const int rem  = y * Wn + x;
    const int p    = b * HWn + rem;
    const int base = b * HWn;

    const int nb = b * 3 * HWn + rem;
    const float ir = image[nb], ig = image[nb + HWn], ib = image[nb + 2 * HWn];
    const float confp  = conf[p];
    const float depthp = depth[p];

    const float* __restrict__ offbase = offs + (size_t)b * 2 * NOFFn * HWn;

    float ccsum = 0.0f, cdsum = 0.0f;

#pragma unroll
    for (int k = 0; k < NOFFn; ++k) {
        const float2 o  = load_off(offbase, offp, OFFP, k, p, rem);
        const float  xb = (float)x - o.x;
        const float  yb = (float)y - o.y;
        float cwk = 0.0f;
        // Reference mask: outside [0,W]x[0,H] -> zero contribution everywhere.
        if (!(xb < 0.0f || xb > (float)Wn || yb < 0.0f || yb > (float)Hn)) {
            const float x0 = floorf(xb), y0 = floorf(yb);
            const float wx = xb - x0,   wy = yb - y0;
            float sd = 0.0f, scp = 0.0f, sr = 0.0f, sg = 0.0f, sb = 0.0f, sc = 0.0f;
#pragma unroll
            for (int c = 0; c < 4; ++c) {
                const float xf = x0 + (float)(c & 1);
                const float yf = y0 + (float)(c >> 1);
                if (xf < 0.0f || xf > (float)(Wn - 1) || yf < 0.0f || yf > (float)(Hn - 1))
                    continue;                       // invalid corner: wgt * valid == 0
                const float wgt = ((c & 1) ? wx : 1.0f - wx) * ((c >> 1) ? wy : 1.0f - wy);
                const int xi = (int)xf;
                const int yi = (int)yf;
                const int qi = yi * Wn + xi;
                const int q  = base + qi;
                const float4 A  = packA[q];          // b128 (L2 hit)
                const float2 oq = load_off(offbase, offp, OFFP, k, q, qi);
                const float den = fmaf(oq.x, A.z, fmaf(oq.y, A.w, A.y)) + 1e-18f;
                const float d1p = fminf(fmaxf(A.x / den, 0.01f), 10.0f);
                const float4 Bp = packB[q];          // b128 (L2 hit)
                const float cq  = conf[q];
                sd  = fmaf(wgt, d1p,  sd);
                scp = fmaf(wgt, Bp.x, scp);
                sr  = fmaf(wgt, Bp.y, sr);
                sg  = fmaf(wgt, Bp.z, sg);
                sb  = fmaf(wgt, Bp.w, sb);
                sc  = fmaf(wgt, cq,   sc);
            }
            const float dr = ir - sr, dg = ig - sg, db = ib - sb;
            const float colw = __expf(-(fmaf(dr, dr, fmaf(dg, dg, db * db))) * (1.0f / 20.0f));
            const float dis  = __fsqrt_rn(fmaf(o.x, o.x, o.y * o.y));
            const float aff  = colw * __expf(-dis * (1.0f / 3.0f));
            const float cc   = scp * aff;
            ccsum += cc;
            cdsum  = fmaf(sd, cc, cdsum);
            cwk    = sc * aff * confp;
        }
        cw[(size_t)k * Npx + p] = cwk;               // coalesced per-k plane
    }

    const float den = ccsum + 10.0f * confp + 1e-16f;        // ALPHA1 = 10
    const float nom = fmaf(10.0f * confp, depthp, cdsum);
    ud_out[p] = fminf(fmaxf(nom / den, 0.1f), 10.0f);
}

// ---------------------------------------------------------------------------
// Pass 2: normal solve. points(q) = ud(q)*Kinv[x_q,y_q,1] is RECOMPUTED at each
// corner from the single gathered scalar ud(q). 2x2 weighted LS per pixel,
// result written with non-temporal stores (never re-read).
// ---------------------------------------------------------------------------
template <bool OFFP>
__global__ void __launch_bounds__(BX * BY)
pass2_kernel(const float* __restrict__ normal, const float* __restrict__ confN,
             const float* __restrict__ K, const float* __restrict__ offs,
             const float2* __restrict__ offp,
             const float* __restrict__ cw, const float* __restrict__ ud,
             float* __restrict__ out) {
    const int x    = blockIdx.x * BX + threadIdx.x;
    const int y    = blockIdx.y * BY + threadIdx.y;
    const int b    = blockIdx.z;
    const int rem  = y * Wn + x;
    const int p    = b * HWn + rem;
    const int base = b * HWn;

    float inv[9];
    kinv3(K + b * 9, inv);                           // scalar loads (uniform b)
    const float v0 = fmaf(inv[0], (float)x, fmaf(inv[1], (float)y, inv[2]));
    const float v1 = fmaf(inv[3], (float)x, fmaf(inv[4], (float)y, inv[5]));
    const float v2 = fmaf(inv[6], (float)x, fmaf(inv[7], (float)y, inv[8]));

    const float u  = ud[p];
    const float Px = u * v0, Py = u * v1, Pz = u * v2;

    const int nb = b * 3 * HWn + rem;
    const float nx0 = normal[nb], ny0 = normal[nb + HWn], nz0 = normal[nb + 2 * HWn];
    const float inz = 1.0f / nz0;
    const float n1x = -nx0 * inz, n1y = -ny0 * inz;

    const float cN = confN[p];
    float A11 = 10.0f * cN, A22 = 10.0f * cN, A12 = 0.0f;    // ALPHA2 = 10
    float b1 = 10.0f * cN * n1x, b2 = 10.0f * cN * n1y;

    const float* __restrict__ offbase = offs + (size_t)b * 2 * NOFFn * HWn;

#pragma unroll
    for (int k = 0; k < NOFFn; ++k) {
        const float cwk = cw[(size_t)k * Npx + p];
        if (cwk == 0.0f) continue;       // masked/zero weight: exact zero contribution
        const float2 o  = load_off(offbase, offp, OFFP, k, p, rem);
        const float  xb = (float)x - o.x;
        const float  yb = (float)y - o.y;
        const float x0 = floorf(xb), y0 = floorf(yb);
        const float wx = xb - x0,   wy = yb - y0;
        float cx = 0.0f, cy = 0.0f, cz = 0.0f;
#pragma unroll
        for (int c = 0; c < 4; ++c) {
            const float xf = x0 + (float)(c & 1);
            const float yf = y0 + (float)(c >> 1);
            if (xf < 0.0f || xf > (float)(Wn - 1) || yf < 0.0f || yf > (float)(Hn - 1))
                continue;
            const float wgt = ((c & 1) ? wx : 1.0f - wx) * ((c >> 1) ? wy : 1.0f - wy);
            const int xi = (int)xf;
            const int yi = (int)yf;
            const float uq = ud[base + yi * Wn + xi];   // single-float gather (L2 hit)
            const float w  = wgt * uq;
            const float q0 = fmaf(inv[0], (float)xi, fmaf(inv[1], (float)yi, inv[2]));
            const float q1 = fmaf(inv[3], (float)xi, fmaf(inv[4], (float)yi, inv[5]));
            const float q2 = fmaf(inv[6], (float)xi, fmaf(inv[7], (float)yi, inv[8]));
            cx = fmaf(w, q0, cx);
            cy = fmaf(w, q1, cy);
            cz = fmaf(w, q2, cz);
        }
        const float rx = Px - cx, ry = Py - cy, rz = Pz - cz;
        A11 = fmaf(cwk * rx, rx, A11);
        A12 = fmaf(cwk * rx, ry, A12);
        A22 = fmaf(cwk * ry, ry, A22);
        b1  = fmaf(cwk * rz, rx, b1);
        b2  = fmaf(cwk * rz, ry, b2);
    }

    const float det  = fmaf(A11, A22, -A12 * A12) + 1e-6f;
    const float idet = 1.0f / det;
    const float n1 = fminf(fmaxf(fmaf(b1, A22, -b2 * A12) * idet, -20.0f), 20.0f);
    const float n2 = fminf(fmaxf(fmaf(b2, A11, -b1 * A12) * idet, -20.0f), 20.0f);
    const float rn = __frsqrt_rn(fmaf(n1, n1, fmaf(n2, n2, 1.0f)));
    // un = -[n1,n2,-1]/||.|| = (-n1, -n2, +1) * rn ; streamed, never re-read ->
    // NT stores keep the L2-resident gather set warm.
    float* unp = out + Npx + nb;
    __builtin_nontemporal_store(-n1 * rn, unp);
    __builtin_nontemporal_store(-n2 * rn, unp + HWn);
    __builtin_nontemporal_store(rn,       unp + 2 * HWn);
}

extern "C" void kernel_launch(void* const* d_in, const int* in_sizes, int n_in,
                              void* d_out, int out_size, void* d_ws, size_t ws_size,
                              hipStream_t stream) {
    const float* depth  = (const float*)d_in[0];   // (B,1,H,W)
    const float* normal = (const float*)d_in[1];   // (B,3,H,W)
    const float* image  = (const float*)d_in[2];   // (B,3,H,W)
    const float* conf   = (const float*)d_in[3];   // (B,1,H,W)
    const float* confN  = (const float*)d_in[4];   // (B,1,H,W)
    const float* K      = (const float*)d_in[5];   // (B,3,3)
    const float* offs   = (const float*)d_in[6];   // (B,16,H,W)

    float* out = (float*)d_out;                    // [updated_depth | un], fp32

    // Workspace layout (16B aligned):
    //   packA : Npx float4                 19.7 MB
    //   packB : Npx float4                 19.7 MB
    //   cw    : NOFF * Npx float           39.3 MB
    //   offp  : NOFF * Npx float2 (opt.)   78.6 MB  -> total 157.3 MB
    float4* packA = (float4*)d_ws;
    float4* packB = packA + Npx;
    float*  cwbuf = (float*)(packB + Npx);
    float2* offpb = (float2*)(cwbuf + (size_t)NOFFn * Npx);

    const size_t baseBytes = (size_t)Npx * 64;          // packA+packB+cw
    const size_t fullBytes = baseBytes + (size_t)Npx * 64;
    const bool useOffp = (ws_size >= fullBytes);        // constant per run

    dim3 block(BX, BY, 1);                 // 256 threads = 8 wave32 waves
    dim3 grid(Wn / BX, Hn / BY, Bn);       // 10 x 120 x 4, exact cover

    if (useOffp) {
        pack_kernel<true><<<grid, block, 0, stream>>>(depth, normal, image, conf, confN,
                                                      K, offs, packA, packB, offpb);
        pass1_kernel<true><<<grid, block, 0, stream>>>(depth, image, conf, offs, offpb,
                                                       packA, packB, cwbuf, out);
        pass2_kernel<true><<<grid, block, 0, stream>>>(normal, confN, K, offs, offpb,
                                                       cwbuf, out, out);
    } else {
        pack_kernel<false><<<grid, block, 0, stream>>>(depth, normal, image, conf, confN,
                                                       K, offs, packA, packB, offpb);
        pass1_kernel<false><<<grid, block, 0, stream>>>(depth, image, conf, offs, offpb,
                                                        packA, packB, cwbuf, out);
        pass2_kernel<false><<<grid, block, 0, stream>>>(normal, confN, K, offs, offpb,
                                                        cwbuf, out, out);
    }
}